// Encoding_9234179687593
// MI455X (gfx1250) — compile-verified
//
#include <hip/hip_runtime.h>
#include <stdint.h>

// Problem constants (match reference)
#define BATCH 16
#define CCH   512      // channels C
#define NCODE 32       // codewords K
#define NSPAT 4096     // H*W
#define TN    64       // positions per tile
#define SPLIT 16       // workgroups per batch
#define TILES_PER_WG ((NSPAT / TN) / SPLIT)   // 4

// LDS pitches (bank-conflict avoidance, 64 banks of 4B)
#define XP    65       // x tile row pitch (floats): 64 + 1 pad  -> stride 65 mod 64 = 1
#define CWP   516      // codeword row pitch: 512 + 4           -> stride mod 64 = 4 (16 lanes distinct banks)
#define WP    33       // weights row pitch: 32 + 1

typedef float v2f __attribute__((ext_vector_type(2)));
typedef float v8f __attribute__((ext_vector_type(8)));
typedef uint32_t v4u __attribute__((ext_vector_type(4)));
typedef int v8i __attribute__((ext_vector_type(8)));
typedef int v4i __attribute__((ext_vector_type(4)));

__global__ __launch_bounds__(256)
void enc_fused_kernel(const float* __restrict__ x,
                      const float* __restrict__ cw,
                      const float* __restrict__ scale,
                      float* __restrict__ out) {
  // LDS: ~208 KB total (CDNA5 WGP has 320 KB)
  __shared__ float lds_x[CCH * XP];       // x tile, layout [c][n_local], pitch 65
  __shared__ float lds_cw[NCODE * CWP];   // codewords [k][c], pitch 516
  __shared__ float lds_w[TN * WP];        // xc then softmax weights [n_local][k]
  __shared__ float lds_x2[TN];
  __shared__ float lds_c2[NCODE];
  __shared__ float lds_scale[NCODE];
  __shared__ float lds_wsum[NCODE];

  const int tid  = threadIdx.x;
  const int wv   = tid >> 5;         // wave id 0..7
  const int lane = tid & 31;
  const int m    = lane & 15;        // lane row/col within half-wave
  const int half = lane >> 4;        // 0: lanes 0-15, 1: lanes 16-31
  const int b    = blockIdx.y;
  const int spl  = blockIdx.x;

  // ---- Stage codewords into padded LDS (coalesced float4 loads) ----
  for (int i = tid * 4; i < NCODE * CCH; i += 256 * 4) {
    const float4 v = *reinterpret_cast<const float4*>(cw + i);
    const int k = i >> 9;            // /512
    const int c = i & (CCH - 1);
    float* dst = &lds_cw[k * CWP + c];
    dst[0] = v.x; dst[1] = v.y; dst[2] = v.z; dst[3] = v.w;
  }
  __syncthreads();
  if (tid < NCODE) {
    float s = 0.f;
    for (int c = 0; c < CCH; ++c) { const float v = lds_cw[tid * CWP + c]; s += v * v; }
    lds_c2[tid]    = s;
    lds_scale[tid] = scale[tid];
    lds_wsum[tid]  = 0.f;
  }

  // Persistent GEMM2 accumulators: 8 output tiles of 16x16 per wave
  v8f accs[8];
  #pragma unroll
  for (int j = 0; j < 8; ++j) accs[j] = v8f{};

  const uint32_t lds_x_off = (uint32_t)(uintptr_t)(void*)lds_x;

  for (int it = 0; it < TILES_PER_WG; ++it) {
    const int n0 = (spl * TILES_PER_WG + it) * TN;

    __syncthreads();  // previous readers of lds_x done (also covers staging for it==0)

    // ---- TDM: async 2D tile load, 512 rows (c) x 64 cols (n), +1 DWORD pad/row ----
    if (tid < 32) {
      const uint64_t ga = (uint64_t)(uintptr_t)(x + (size_t)b * CCH * NSPAT + n0);
      v4u g0;
      g0[0] = 1u;                                   // count=1 (valid user descriptor)
      g0[1] = lds_x_off;                            // lds_addr (bytes)
      g0[2] = (uint32_t)ga;                         // global_addr lo
      g0[3] = (uint32_t)(ga >> 32) | (2u << 30);    // global_addr hi | type=2 (image)
      v8i g1;
      g1[0] = (2 << 16)      // data_size = 4 bytes
            | (1 << 20)      // pad_enable
            | (5 << 22)      // pad_interval: 64 DWORDs
            | (0 << 25);     // pad_amount: 1 DWORD
      g1[1] = (NSPAT & 0xFFFF) << 16;                    // tensor_dim0 lo16
      g1[2] = (NSPAT >> 16) | ((CCH & 0xFFFF) << 16);    // dim0 hi | tensor_dim1 lo16
      g1[3] = (CCH >> 16) | (TN << 16);                  // dim1 hi | tile_dim0 = 64
      g1[4] = CCH;                                       // tile_dim1 = 512, tile_dim2 = 0
      g1[5] = NSPAT;                                     // tensor_dim0_stride lo32
      g1[6] = 0;
      g1[7] = 0;
      const v4i gz4 = v4i{};
      const v8i gz8 = v8i{};
      __builtin_amdgcn_tensor_load_to_lds(g0, g1, gz4, gz4, gz8, 0);
      __builtin_amdgcn_s_wait_tensorcnt(0);
    }
    __syncthreads();

    // ---- GEMM1: xc[n, k] = sum_c xf[n,c] * cw[k,c]  (fp32 WMMA 16x16x4) ----
    // wave -> one 16x16 output tile: nt in 0..3 (n sub-tile), kh in 0..1 (k half)
    {
      const int nt    = wv & 3;
      const int kh    = wv >> 2;
      const int nbase = nt * 16;
      const int k0    = kh * 16;
      v8f acc = v8f{};
      float sq = 0.f;
      // A[m][kk] = lds_x[(c+kk)*XP + nbase + m], kk = 2*half + {0,1}
      const float* xa = &lds_x[(2 * half) * XP + nbase + m];
      // B[kk][j=m] = lds_cw[(k0+m)*CWP + c + kk]
      const float* ba = &lds_cw[(k0 + m) * CWP + 2 * half];
      #pragma unroll 4
      for (int c = 0; c < CCH; c += 4) {
        v2f a, bb;
        a.x  = xa[c * XP];
        a.y  = xa[c * XP + XP];
        bb.x = ba[c];
        bb.y = ba[c + 1];
        acc = __builtin_amdgcn_wmma_f32_16x16x4_f32(
            false, a, false, bb, (short)0, acc, false, false);
        sq += a.x * a.x + a.y * a.y;
      }
      // x2[n]: lanes L and L^16 cover complementary c's of row nbase+m
      const float tot = sq + __shfl_xor(sq, 16, 32);
      if (kh == 0 && half == 0) lds_x2[nbase + m] = tot;
      // D[M=v+8*half][N=m] -> xc row n = nbase+v+8*half, col k = k0+m
      #pragma unroll
      for (int v = 0; v < 8; ++v)
        lds_w[(nbase + v + 8 * half) * WP + k0 + m] = acc[v];
    }
    __syncthreads();

    // ---- softmax over K=32 per position (one thread per row) ----
    if (tid < TN) {
      const float x2v = lds_x2[tid];
      float mx = -3.4e38f;
      for (int k = 0; k < NCODE; ++k) {
        const float sv = lds_scale[k] * (x2v - 2.f * lds_w[tid * WP + k] + lds_c2[k]);
        mx = fmaxf(mx, sv);
      }
      float sum = 0.f;
      for (int k = 0; k < NCODE; ++k) {
        const float sv = lds_scale[k] * (x2v - 2.f * lds_w[tid * WP + k] + lds_c2[k]);
        const float e = __expf(sv - mx);
        lds_w[tid * WP + k] = e;   // safe: position k not re-read after write
        sum += e;
      }
      const float inv = 1.f / sum;
      for (int k = 0; k < NCODE; ++k) lds_w[tid * WP + k] *= inv;
    }
    __syncthreads();

    // wsum[k] += sum_n w[n,k] (wave 0 only; read-read overlap with GEMM2 is fine)
    if (tid < NCODE) {
      float s = 0.f;
      for (int n = 0; n < TN; ++n) s += lds_w[n * WP + tid];
      lds_wsum[tid] += s;
    }

    // ---- GEMM2: P[k, c] += sum_n w[n,k] * xf[n,c]  (accumulate in registers) ----
    #pragma unroll
    for (int j = 0; j < 8; ++j) {
      const int t  = wv + j * 8;       // 64 tiles: kh = t&1, cc = t>>1
      const int kh = t & 1;
      const int c0 = (t >> 1) * 16;
      v8f acc2 = accs[j];
      // A[m=k][kk=n] = lds_w[(nb+kk)*WP + kh*16 + m]
      const float* wa = &lds_w[(2 * half) * WP + kh * 16 + m];
      // B[kk=n][j=c] = lds_x[(c0+m)*XP + nb + kk]
      const float* xb = &lds_x[(c0 + m) * XP + 2 * half];
      #pragma unroll 4
      for (int nb = 0; nb < TN; nb += 4) {
        v2f a, bb;
        a.x  = wa[nb * WP];
        a.y  = wa[nb * WP + WP];
        bb.x = xb[nb];
        bb.y = xb[nb + 1];
        acc2 = __builtin_amdgcn_wmma_f32_16x16x4_f32(
            false, a, false, bb, (short)0, acc2, false, false);
      }
      accs[j] = acc2;
    }
  }

  __syncthreads();  // lds_wsum final

  // ---- final: out[b,k,c] += P[k,c] - wsum[k]*cw[k,c]  (one atomic pass) ----
  const size_t outb = (size_t)b * NCODE * CCH;
  #pragma unroll
  for (int j = 0; j < 8; ++j) {
    const int t  = wv + j * 8;
    const int kh = t & 1;
    const int c0 = (t >> 1) * 16;
    #pragma unroll
    for (int v = 0; v < 8; ++v) {
      const int k = kh * 16 + v + 8 * half;
      const int c = c0 + m;
      const float val = accs[j][v] - lds_wsum[k] * lds_cw[k * CWP + c];
      atomicAdd(out + outb + (size_t)k * CCH + c, val);
    }
  }
}

extern "C" void kernel_launch(void* const* d_in, const int* in_sizes, int n_in,
                              void* d_out, int out_size, void* d_ws, size_t ws_size,
                              hipStream_t stream) {
  (void)in_sizes; (void)n_in; (void)d_ws; (void)ws_size;
  const float* x     = (const float*)d_in[0];
  const float* cw    = (const float*)d_in[1];
  const float* scale = (const float*)d_in[2];
  float* out = (float*)d_out;

  (void)hipMemsetAsync(out, 0, (size_t)out_size * sizeof(float), stream);

  dim3 grid(SPLIT, BATCH);
  enc_fused_kernel<<<grid, 256, 0, stream>>>(x, cw, scale, out);
}